// MPM_11579231830686
// MI455X (gfx1250) — compile-verified
//
#include <hip/hip_runtime.h>
#include <cstdint>
#include <math.h>

// ---------------- MPM constants (must match reference) ----------------
#define NPART   3000
#define NGX     120
#define NGY     80
#define NG      (NGX*NGY)          // 9600 grid nodes
#define DXC     4.0f
#define INV_DXC 0.25f
#define DTC     0.05f
#define ALPHAC  (-0.1f)
#define NSUB    2
#define BLOCK   1024               // 32 wave32s -> one WGP, single workgroup
#define PPT     3                  // ceil(3000/1024) particles per thread
#define GPT     10                 // ceil(9600/1024) grid nodes per thread

// CDNA5 async LDS->global store path (ASYNCcnt-tracked).
// Probe round 1 confirmed the builtin exists with signature
//   void(v2i addrspace(1)*, v2i addrspace(3)*, imm int offset, imm int cpol)
#if defined(__gfx1250__) && __has_builtin(__builtin_amdgcn_global_store_async_from_lds_b64)
#define USE_ASYNC 1
#else
#define USE_ASYNC 0
#endif

typedef int v2i_t __attribute__((vector_size(2 * sizeof(int))));
typedef __attribute__((address_space(1))) v2i_t* gptr_v2i;
typedef __attribute__((address_space(3))) v2i_t* lptr_v2i;

__device__ __forceinline__ float wfn(float dis) {
    // quadratic B-spline window, support |dis*INV_DX| < 1.5
    float nd = fabsf(dis * INV_DXC);
    float t  = 1.5f - nd;
    return (nd < 0.5f) ? (0.75f - nd * nd)
                       : ((nd < 1.5f) ? (0.5f * t * t) : 0.0f);
}

__global__ __launch_bounds__(BLOCK)
void mpm_persistent_kernel(const float* __restrict__ pos_in,
                           const float* __restrict__ vel_in,
                           const float* __restrict__ C_in,
                           const float* __restrict__ J_in,
                           const float* __restrict__ E_in,
                           const float* __restrict__ center_in,
                           float* __restrict__ out)
{
    // Whole grid lives in LDS: gv[2*NG] (vx,vy interleaved) + gm[NG] = 115.2 KB.
    // Only possible because CDNA5 gives 320 KB LDS per WGP.
    extern __shared__ float smem[];
    float* gv = smem;             // 2*NG floats
    float* gm = smem + 2 * NG;    // NG floats

    const int tid = threadIdx.x;

    // ---- particle state lives in registers across both substeps ----
    float px[PPT], py[PPT], vx[PPT], vy[PPT];
    float c00[PPT], c01[PPT], c10[PPT], c11[PPT], Jp[PPT];
    bool  act[PPT];
    for (int k = 0; k < PPT; ++k) {
        int p  = tid + k * BLOCK;
        act[k] = (p < NPART);
        int q  = act[k] ? p : 0;
        px[k]  = pos_in[2 * q];     py[k]  = pos_in[2 * q + 1];
        vx[k]  = vel_in[2 * q];     vy[k]  = vel_in[2 * q + 1];
        c00[k] = C_in[4 * q];       c01[k] = C_in[4 * q + 1];
        c10[k] = C_in[4 * q + 2];   c11[k] = C_in[4 * q + 3];
        Jp[k]  = J_in[q];
    }
    const float E0  = E_in[0];
    const float ccx = center_in[0], ccy = center_in[1];

    for (int s = 0; s < NSUB; ++s) {
#if USE_ASYNC
        // Previous substep's async LDS->global stores must finish before we
        // re-zero gv. Each wave drains its own ASYNCcnt, then the workgroup
        // barrier below makes that global across waves.
#if __has_builtin(__builtin_amdgcn_s_wait_asynccnt)
        __builtin_amdgcn_s_wait_asynccnt(0);
#else
        asm volatile("s_wait_asynccnt 0" ::: "memory");
#endif
#endif
        __syncthreads();

        // ---- zero grid ----
        for (int k = 0; k < GPT; ++k) {
            int g = tid + k * BLOCK;
            if (g < NG) { gv[2 * g] = 0.f; gv[2 * g + 1] = 0.f; gm[g] = 0.f; }
        }
        __syncthreads();

        // ---- P2G scatter (ds_add_f32 atomics into LDS) ----
        for (int k = 0; k < PPT; ++k) if (act[k]) {
            // tangential-velocity external force: -DT * m * v_t^2 / r^2 * r_vec
            float rx = px[k] - ccx, ry = py[k] - ccy;
            float r2 = rx * rx + ry * ry;
            float dp = (rx * vx[k] + ry * vy[k]) / r2;
            float tx = vx[k] - dp * rx, ty = vy[k] - dp * ry;
            float vt2 = tx * tx + ty * ty;
            float fs  = -DTC * vt2 / r2;            // P_MASS = 1
            float mvx = vx[k] + fs * rx;            // P_MASS*vel + ext_force
            float mvy = vy[k] + fs * ry;
            float stress = -DTC * 4.0f * INV_DXC * INV_DXC * E0 * (Jp[k] - 1.0f);

            int bi = (int)floorf(px[k] * INV_DXC - 0.5f);
            int bj = (int)floorf(py[k] * INV_DXC - 0.5f);
            float dxv[3], dyv[3], wxv[3], wyv[3];
            #pragma unroll
            for (int d = 0; d < 3; ++d) {
                dxv[d] = (float)(bi + d) * DXC - px[k]; wxv[d] = wfn(dxv[d]);
                dyv[d] = (float)(bj + d) * DXC - py[k]; wyv[d] = wfn(dyv[d]);
            }
            #pragma unroll
            for (int a = 0; a < 3; ++a) {
                int gi = bi + a;
                if (gi < 0 || gi >= NGX) continue;
                #pragma unroll
                for (int b = 0; b < 3; ++b) {
                    int gj = bj + b;
                    if (gj < 0 || gj >= NGY) continue;
                    float w  = wxv[a] * wyv[b];
                    float dx = dxv[a], dy = dyv[b];
                    int   g  = gi * NGY + gj;
                    float cx = w * (mvx + c00[k] * dx + c01[k] * dy + stress * dx);
                    float cy = w * (mvy + c10[k] * dx + c11[k] * dy + stress * dy);
                    atomicAdd(&gv[2 * g],     cx);
                    atomicAdd(&gv[2 * g + 1], cy);
                    atomicAdd(&gm[g],         w);
                }
            }
        }
        __syncthreads();

        // ---- grid normalize: gv := grid_v_out ----
        for (int k = 0; k < GPT; ++k) {
            int g = tid + k * BLOCK;
            if (g < NG) {
                float m = gm[g];
                if (m > 0.f) { float inv = 1.0f / m; gv[2*g] *= inv; gv[2*g+1] *= inv; }
                else         { gv[2*g] = 0.f; gv[2*g+1] = 0.f; }
            }
        }
        __syncthreads();

        // ---- stream grid_v_out field to HBM straight from LDS (async),
        //      overlapped with the G2P gather below ----
        float* outp = out + (size_t)s * (2 * NG);
        for (int k = 0; k < GPT; ++k) {
            int g = tid + k * BLOCK;
            if (g < NG) {
#if USE_ASYNC
                __builtin_amdgcn_global_store_async_from_lds_b64(
                    (gptr_v2i)(uintptr_t)(outp + 2 * g),
                    (lptr_v2i)(uintptr_t)(&gv[2 * g]),
                    0, 0);
#else
                outp[2 * g]     = gv[2 * g];
                outp[2 * g + 1] = gv[2 * g + 1];
#endif
            }
        }

        // ---- G2P gather + particle advection + boundaries ----
        for (int k = 0; k < PPT; ++k) if (act[k]) {
            int bi = (int)floorf(px[k] * INV_DXC - 0.5f);
            int bj = (int)floorf(py[k] * INV_DXC - 0.5f);
            float dxv[3], dyv[3], wxv[3], wyv[3];
            #pragma unroll
            for (int d = 0; d < 3; ++d) {
                dxv[d] = (float)(bi + d) * DXC - px[k]; wxv[d] = wfn(dxv[d]);
                dyv[d] = (float)(bj + d) * DXC - py[k]; wyv[d] = wfn(dyv[d]);
            }
            float nvx = 0.f, nvy = 0.f, n00 = 0.f, n01 = 0.f, n10 = 0.f, n11 = 0.f;
            #pragma unroll
            for (int a = 0; a < 3; ++a) {
                int gi = bi + a;
                if (gi < 0 || gi >= NGX) continue;
                #pragma unroll
                for (int b = 0; b < 3; ++b) {
                    int gj = bj + b;
                    if (gj < 0 || gj >= NGY) continue;
                    float w  = wxv[a] * wyv[b];
                    float dx = dxv[a], dy = dyv[b];
                    int   g  = gi * NGY + gj;
                    float gx = gv[2 * g], gy = gv[2 * g + 1];
                    nvx += w * gx;       nvy += w * gy;
                    n00 += w * gx * dx;  n01 += w * gx * dy;
                    n10 += w * gy * dx;  n11 += w * gy * dy;
                }
            }
            const float kk = 4.0f * INV_DXC * INV_DXC;
            n00 *= kk; n01 *= kk; n10 *= kk; n11 *= kk;
            Jp[k] *= (1.0f + DTC * (n00 + n11));
            float X = px[k] + DTC * nvx, Y = py[k] + DTC * nvy;
            float VX = nvx, VY = nvy;
            // apply_boundary, sequential, same order as reference
            if (Y < 10.f  && VY < 0.f) { Y = 10.f;  VY *= ALPHAC; }
            if (Y > 302.f && VY > 0.f) { Y = 302.f; VY *= ALPHAC; }
            if (X > 470.f && VX > 0.f) { X = 470.f; VX *= ALPHAC; }
            if (X < 10.f  && VX < 0.f) { X = 10.f;  VX *= ALPHAC; }
            if (Y > 138.f && Y <= 160.f && X >= 280.f && X <= 320.f && VY > 0.f) { Y = 138.f; VY *= ALPHAC; }
            if (Y > 160.f && Y < 182.f  && X >= 280.f && X <= 320.f && VY < 0.f) { Y = 182.f; VY *= ALPHAC; }
            if (X >= 300.f && X < 322.f && Y >= 140.f && Y <= 180.f && VX < 0.f) { X = 322.f; VX *= ALPHAC; }
            if (X > 278.f  && X < 300.f && Y >= 140.f && Y <= 180.f && VX > 0.f) { X = 278.f; VX *= ALPHAC; }
            px[k] = X; py[k] = Y; vx[k] = VX; vy[k] = VY;
            c00[k] = n00; c01[k] = n01; c10[k] = n10; c11[k] = n11;
        }
        // loop-top wait + barrier fences the async stores before re-zeroing;
        // S_ENDPGM performs an implicit wait-idle for the final substep.
    }
}

extern "C" void kernel_launch(void* const* d_in, const int* in_sizes, int n_in,
                              void* d_out, int out_size, void* d_ws, size_t ws_size,
                              hipStream_t stream) {
    (void)in_sizes; (void)n_in; (void)out_size; (void)d_ws; (void)ws_size;
    const float* pos    = (const float*)d_in[0];
    const float* vel    = (const float*)d_in[1];
    const float* C      = (const float*)d_in[2];
    const float* J      = (const float*)d_in[3];
    const float* E      = (const float*)d_in[4];
    const float* center = (const float*)d_in[5];
    // d_in[6] (grid_pos) is implicit: node (i,j) sits at (i*DX, j*DX) exactly.

    const size_t shmem = (size_t)(3 * NG) * sizeof(float);   // 115,200 B < 320 KB LDS/WGP
    mpm_persistent_kernel<<<dim3(1), dim3(BLOCK), shmem, stream>>>(
        pos, vel, C, J, E, center, (float*)d_out);
}